// HybridCalibrationModel_32023276159622
// MI455X (gfx1250) — compile-verified
//
#include <hip/hip_runtime.h>

#define ROWS 16
#define LROW (1 << 21)            // 2097152
#define TAPS 63
#define CTAPS 66                  // fused 66-tap filter
#define WG_OUT 2048               // outputs per workgroup (8 waves x 256)
#define STAGE 2116                // WG_OUT + 68 halo
#define LDSX 2120                 // staged x floats (rounded to float4)
#define LDSC 112                  // comb table with zero guards, offset +16

typedef __attribute__((ext_vector_type(2))) float v2f;
typedef __attribute__((ext_vector_type(8))) float v8f;

// ---------------------------------------------------------------------------
// Kernel 1: build fused 66-tap filter comb[j] = gain * sum_k fw[k]*fir[j-k]
// fw = Farrow cubic weights for fixed fractional delay d.
// ---------------------------------------------------------------------------
__global__ void comb_kernel(const float* __restrict__ gain_p,
                            const float* __restrict__ delay_p,
                            const float* __restrict__ fir,
                            float* __restrict__ comb) {
    int j = threadIdx.x;
    if (j >= CTAPS) return;
    float d = *delay_p, g = *gain_p;
    float d2 = d * d, d3 = d2 * d;
    float w[4];
    w[0] = -d * (1.f / 3.f) + d2 * 0.5f - d3 * (1.f / 6.f);
    w[1] = 1.f - d * 0.5f - d2 + d3 * 0.5f;
    w[2] = d + d2 * 0.5f - d3 * 0.5f;
    w[3] = -d * (1.f / 6.f) + d3 * (1.f / 6.f);
    float c = 0.f;
#pragma unroll
    for (int k = 0; k < 4; ++k) {
        int t = j - k;
        if (t >= 0 && t < TAPS) c += w[k] * fir[t];
    }
    comb[j] = g * c;
}

// ---------------------------------------------------------------------------
// Kernel 2: fused 66-tap FIR via V_WMMA_F32_16X16X4_F32 (Toeplitz mapping).
// Each wave computes 256 contiguous outputs: D[m,n] = y[tb + 16m + n],
// A[m,k] = x[tb - 32 + 16m + k] (k in [0,84)), B[k,n] = comb[k-n].
// ---------------------------------------------------------------------------
__device__ __forceinline__ float ld_clamp(const float* x, long long g, long long N) {
    long long t = g < 0 ? 0 : (g >= N ? N - 1 : g);
    return x[t];
}

__global__ __launch_bounds__(256) void fir_wmma_kernel(const float* __restrict__ x,
                                                       const float* __restrict__ comb,
                                                       float* __restrict__ out) {
    __shared__ float lx[LDSX];
    __shared__ float lc[LDSC];

    const long long N = (long long)ROWS * LROW;
    const long long wg_base = (long long)blockIdx.x * WG_OUT;
    const int tid = threadIdx.x;

    // Stage zero-guarded comb table: lc[16 + j] = comb[j] for j in [0,66)
    if (tid < LDSC) lc[tid] = (tid >= 16 && tid < 16 + CTAPS) ? comb[tid - 16] : 0.f;

    // Stage x[wg_base-32 .. wg_base+2087] into LDS as float4 (64B aligned).
    const long long g0 = wg_base - 32;
#pragma unroll 2
    for (int v = tid; v < LDSX / 4; v += 256) {
        long long g = g0 + 4LL * v;
        float4 val;
        if (g >= 0 && g + 3 < N) {
            val = *(const float4*)(x + g);
        } else {
            val.x = ld_clamp(x, g + 0, N);
            val.y = ld_clamp(x, g + 1, N);
            val.z = ld_clamp(x, g + 2, N);
            val.w = ld_clamp(x, g + 3, N);
        }
        *(float4*)(lx + 4 * v) = val;
    }
    __syncthreads();

    const int w    = tid >> 5;        // wave id (wave32)
    const int lane = tid & 31;
    const int mn   = lane & 15;       // row m for A, col n for B/D
    const int kh   = lane >> 4;       // K-half selector (K pairs 0-1 vs 2-3)

    const float* ax = lx + 256 * w + 16 * mn + 2 * kh;   // even -> 8B aligned
    const float* bx = lc + 16 + 2 * kh - mn;             // indices in [1,99]

    v8f c = {0.f, 0.f, 0.f, 0.f, 0.f, 0.f, 0.f, 0.f};
#pragma unroll
    for (int k0 = 0; k0 < 84; k0 += 4) {
        v2f a = *(const v2f*)(ax + k0);
        v2f b;
        b[0] = bx[k0];
        b[1] = bx[k0 + 1];
        c = __builtin_amdgcn_wmma_f32_16x16x4_f32(
                /*neg_a=*/false, a, /*neg_b=*/false, b,
                /*c_mod=*/(short)0, c, /*reuse_a=*/false, /*reuse_b=*/false);
    }

    // D layout: VGPR r holds rows M=r (lanes 0-15) and M=r+8 (lanes 16-31).
    const long long tb = wg_base + 256LL * w;
    float* op = out + tb + mn;
#pragma unroll
    for (int r = 0; r < 8; ++r) op[16 * (r + 8 * kh)] = c[r];
}

// ---------------------------------------------------------------------------
// Kernel 3: exact recompute of first/last 64 outputs of each row
// (Farrow edge-clamp + FIR zero-padding semantics).
// ---------------------------------------------------------------------------
__global__ void edge_kernel(const float* __restrict__ x,
                            const float* __restrict__ gain_p,
                            const float* __restrict__ delay_p,
                            const float* __restrict__ fir,
                            float* __restrict__ out) {
    const int L = LROW;
    const int row = blockIdx.x;
    const int tid = threadIdx.x;
    const long long base = (long long)row * L;
    const int p = (tid < 64) ? tid : (L - 128 + tid);

    float d = *delay_p, g = *gain_p;
    float d2 = d * d, d3 = d2 * d;
    float w0 = -d * (1.f / 3.f) + d2 * 0.5f - d3 * (1.f / 6.f);
    float w1 = 1.f - d * 0.5f - d2 + d3 * 0.5f;
    float w2 = d + d2 * 0.5f - d3 * 0.5f;
    float w3 = -d * (1.f / 6.f) + d3 * (1.f / 6.f);

    const float* xr = x + base;
    float acc = 0.f;
    for (int t = 0; t < TAPS; ++t) {
        int n = p + t - TAPS / 2;
        if (n < 0 || n >= L) continue;                 // FIR zero padding
        int i0 = n - 1 < 0 ? 0 : n - 1;                // Farrow edge clamp
        int i2 = n + 1 > L - 1 ? L - 1 : n + 1;
        int i3 = n + 2 > L - 1 ? L - 1 : n + 2;
        float s0 = xr[i0] * g, s1 = xr[n] * g, s2 = xr[i2] * g, s3 = xr[i3] * g;
        float del = w0 * s0 + w1 * s1 + w2 * s2 + w3 * s3;
        acc += fir[t] * del;
    }
    out[base + p] = acc;
}

// ---------------------------------------------------------------------------
extern "C" void kernel_launch(void* const* d_in, const int* in_sizes, int n_in,
                              void* d_out, int out_size, void* d_ws, size_t ws_size,
                              hipStream_t stream) {
    const float* x     = (const float*)d_in[0];  // (16,1,2097152) f32
    const float* gain  = (const float*)d_in[1];  // scalar
    const float* delay = (const float*)d_in[2];  // scalar
    const float* fir   = (const float*)d_in[3];  // 63 taps
    float* out  = (float*)d_out;
    float* comb = (float*)d_ws;                  // 66 floats of scratch

    // 1) fused filter taps
    comb_kernel<<<1, 96, 0, stream>>>(gain, delay, fir, comb);

    // 2) bulk fused FIR via WMMA: 16*2^21 outputs / 2048 per block
    const int nblocks = (ROWS * LROW) / WG_OUT;  // 16384
    fir_wmma_kernel<<<nblocks, 256, 0, stream>>>(x, comb, out);

    // 3) exact edges (overwrites 64 samples at each end of each row)
    edge_kernel<<<ROWS, 128, 0, stream>>>(x, gain, delay, fir, out);
}